// HierarchicalDiffusion_46377056863084
// MI455X (gfx1250) — compile-verified
//
#include <hip/hip_runtime.h>

// ---------------------------------------------------------------------------
// HierarchicalDiffusion mix for MI455X (gfx1250).
//
// out = prs * (1-t);  out[V:V+C] += (segsum(prs[:V-1], cdict)*A + Bc) * c_t;
// out[MASK] += m_t;   with c_t = -(1-t)ln(1-t), m_t = t + (1-t)ln(1-t).
//
// Bandwidth-bound: ~1.66 GB total -> ~71 us floor @ 23.3 TB/s. WMMA is the
// wrong tool (one-hot GEMM adds ~13 GB of L2 traffic for junk FLOPs); the
// right CDNA5 paths are ds_add_f32 segment sums in LDS and the ASYNCcnt
// global->LDS DMA pipeline (global_load_async_to_lds_b32 + s_wait_asynccnt),
// with non-temporal hints on the once-read/once-written stream so the 201 KB
// cluster_dict stays resident in the 192 MB L2 for all 4096 blocks.
// ---------------------------------------------------------------------------

#define V_SZ    50257
#define C_SZ    512
#define VC      (V_SZ + C_SZ)     // 50769 floats per token row
#define MASK_ID (V_SZ - 1)        // 50256
#define NVOC    (V_SZ - 1)        // 50256 entries feed the segment sum
#define CH      2048              // chunk size (floats) staged through LDS
#define TPB     512               // 16 wave32 per workgroup

// Issue one chunk of the row into an LDS staging buffer via async DMA.
// Always issues exactly 4 async b32 ops per wave (tail lanes clamp the
// global address instead of predicating) so ASYNCcnt bookkeeping is exact.
// th:TH_LOAD_NT -> the vocab stream is read once; don't churn L2 with it.
__device__ __forceinline__ void issue_chunk(const float* __restrict__ row,
                                            int base, float* buf) {
  const int tid = threadIdx.x;
#pragma unroll
  for (int j = 0; j < 4; ++j) {
    int v = base + tid + j * TPB;
    int vclamp = (v < NVOC) ? v : (NVOC - 1);
    // Low 32 bits of a generic pointer to LDS == byte offset within LDS.
    unsigned loff = (unsigned)(unsigned long long)(const void*)&buf[tid + j * TPB];
    const float* gp = row + vclamp;
    asm volatile("global_load_async_to_lds_b32 %0, %1, off th:TH_LOAD_NT"
                 :: "v"(loff), "v"(gp)
                 : "memory");
  }
}

__global__ __launch_bounds__(TPB) void hier_diff_kernel(
    const float* __restrict__ prs, const float* __restrict__ t,
    const int* __restrict__ cdict, float* __restrict__ out) {
  __shared__ float s_cl[C_SZ];       // 2 KB cluster accumulators
  __shared__ float s_buf[2][CH];     // 16 KB double-buffered staging

  const int tid = threadIdx.x;
  const int b   = blockIdx.y;                       // batch index, no division
  const int tok = b * gridDim.x + blockIdx.x;       // flat token index

  const float tb    = t[b];
  const float alpha = 1.0f - tb;                  // alpha_t (gamma == 1)
  const float lg    = logf(alpha);
  const float c_t   = -alpha * lg;                // cluster mass
  const float m_t   = tb + alpha * lg;            // mask mass

  const size_t rbase = (size_t)tok * (size_t)VC;
  const float* __restrict__ row  = prs + rbase;
  float* __restrict__       orow = out + rbase;

  // Zero cluster accumulators; kick off DMA of chunk 0 before the barrier.
  for (int c = tid; c < C_SZ; c += TPB) s_cl[c] = 0.0f;
  issue_chunk(row, 0, &s_buf[0][0]);
  __syncthreads();

  const int nch = (NVOC + CH - 1) / CH;            // 25 chunks (24 full + 1104)
  for (int k = 0; k < nch; ++k) {
    if (k + 1 < nch) {
      issue_chunk(row, (k + 1) * CH, &s_buf[(k + 1) & 1][0]);
      // 8 outstanding (chunks k, k+1); retire chunk k, keep k+1 in flight.
      asm volatile("s_wait_asynccnt 4" ::: "memory");
    } else {
      asm volatile("s_wait_asynccnt 0" ::: "memory");
    }
    const int base = k * CH;
    const float* sb = &s_buf[k & 1][0];
#pragma unroll
    for (int j = 0; j < 4; ++j) {
      int v = base + tid + j * TPB;
      if (v < NVOC) {
        float p = sb[tid + j * TPB];                       // ds_load_b32
        __builtin_nontemporal_store(p * alpha, orow + v);  // streaming store
        atomicAdd(&s_cl[cdict[v]], p);                     // ds_add_f32
      }
    }
  }

  __syncthreads();

  // Cluster tail: out[V+c] = prs[V+c]*alpha + (segsum*A + Bc)*c_t
  const float Aff = 1.0f - 0.1f * (1.0f + 1.0f / (float)C_SZ);
  const float Bff = 0.1f / (float)C_SZ;
  for (int c = tid; c < C_SZ; c += TPB) {
    float pc = s_cl[c] * Aff + Bff;
    float pv = __builtin_nontemporal_load(row + V_SZ + c);
    __builtin_nontemporal_store(pv * alpha + pc * c_t, orow + V_SZ + c);
  }
  // Mask slot (index V-1, excluded from the segment sum).
  if (tid == 0) {
    float pm = __builtin_nontemporal_load(row + MASK_ID);
    __builtin_nontemporal_store(pm * alpha + m_t, orow + MASK_ID);
  }
}

extern "C" void kernel_launch(void* const* d_in, const int* in_sizes, int n_in,
                              void* d_out, int out_size, void* d_ws, size_t ws_size,
                              hipStream_t stream) {
  const float* prs   = (const float*)d_in[0];
  const float* t     = (const float*)d_in[1];
  const int*   cdict = (const int*)d_in[2];
  float*       out   = (float*)d_out;

  const int B      = in_sizes[1];            // t has B elements
  const int tokens = in_sizes[0] / VC;       // B * S
  const int S      = tokens / B;

  hipLaunchKernelGGL(hier_diff_kernel, dim3(S, B), dim3(TPB), 0, stream,
                     prs, t, cdict, out);
}